// GCN2_515396076078
// MI455X (gfx1250) — compile-verified
//
#include <hip/hip_runtime.h>

#define N_NODES 50000
#define N_EDGES 800000
#define IN_DIM  128
#define HID     64
#define OUT_DIM 10
#define ROW_BLOCKS (N_NODES / 16)   // 3125, exact

typedef __attribute__((ext_vector_type(2))) float v2f;
typedef __attribute__((ext_vector_type(8))) float v8f;

// ---------------- degree / normalization ----------------
__global__ void k_init_deg(float* deg) {
    int i = blockIdx.x * blockDim.x + threadIdx.x;
    if (i < N_NODES) deg[i] = 1.0f;                 // implicit self-loop
}

__global__ void k_edge_deg(const int* __restrict__ dst, float* deg) {
    int e = blockIdx.x * blockDim.x + threadIdx.x;
    if (e < N_EDGES) atomicAdd(&deg[dst[e]], 1.0f);
}

__global__ void k_rsqrt_inplace(float* deg) {
    int i = blockIdx.x * blockDim.x + threadIdx.x;
    if (i < N_NODES) deg[i] = rsqrtf(deg[i]);       // deg^{-1/2}
}

// ---- fused GCN layer GEMM: H = (relu?)A @ W ; AGG = H*dinv^2 + bias ----
// W staged once per block into LDS, pair-swizzled: (k,n) -> [(k>>1)*PSTRIDE + 2n + (k&1)]
// so each B fragment (K pair, fixed n) is one contiguous ds_load_b64.
// PSTRIDE = 160 floats (= 32 mod 64 banks): half-wave 0 reads pair-row r
// (banks 0..31), half-wave 1 reads pair-row r+1 (banks 32..63) -> conflict-free.
// B address = loop-invariant lane base + compile-time constant per unrolled step,
// so the ds_load uses its 16-bit immediate offset (max ~40KB, fits).
// In-place A==AGG is safe: a wave reads only its own 16 rows, and those loads are
// data-dependency ordered before the epilogue stores.
template <int K, bool RELU>
__global__ __launch_bounds__(256) void k_gcn_gemm(const float* __restrict__ A,
                                                  const float* __restrict__ W,
                                                  const float* __restrict__ bias,
                                                  const float* __restrict__ dinv,
                                                  float* __restrict__ H,
                                                  float* __restrict__ AGG) {
    constexpr int PSTRIDE = 2 * HID + 32;           // 160 floats per pair-row
    __shared__ float Wl[(K / 2) * PSTRIDE];

    // stage W (K x 64) into LDS, swizzled; all 256 threads participate
    for (int i = threadIdx.x; i < K * HID; i += 256) {
        int k = i >> 6;                             // HID == 64
        int n = i & 63;
        Wl[(k >> 1) * PSTRIDE + (n << 1) + (k & 1)] = W[i];
    }
    __syncthreads();

    const int lane   = threadIdx.x & 31;
    const int wave   = threadIdx.x >> 5;
    const int rowBlk = blockIdx.x * 8 + wave;
    if (rowBlk < ROW_BLOCKS) {                      // wave-uniform guard, no early return
        const int rowBase = rowBlk * 16;
        const int m    = lane & 15;                 // A row within tile
        const int n    = lane & 15;                 // B/C col within tile
        const int half = lane >> 4;                 // 0 or 1
        const int kOff = half * 2;                  // K sub-offset per half-wave

        v8f c0 = {}, c1 = {}, c2 = {}, c3 = {};
        const float* arow  = A + (size_t)(rowBase + m) * K + kOff;
        const float* wbase = &Wl[half * PSTRIDE + 2 * n];   // loop-invariant base

#pragma unroll
        for (int k0 = 0; k0 < K; k0 += 4) {
            float a0 = arow[k0];
            float a1 = arow[k0 + 1];
            if (RELU) { a0 = fmaxf(a0, 0.0f); a1 = fmaxf(a1, 0.0f); }
            v2f a; a[0] = a0; a[1] = a1;

            const float* wp = wbase + (k0 >> 1) * PSTRIDE;  // constant offset per step
            v2f b0  = *(const v2f*)(wp);
            v2f b1v = *(const v2f*)(wp + 32);
            v2f b2v = *(const v2f*)(wp + 64);
            v2f b3v = *(const v2f*)(wp + 96);

            c0 = __builtin_amdgcn_wmma_f32_16x16x4_f32(false, a, false, b0,  (short)0, c0, false, false);
            c1 = __builtin_amdgcn_wmma_f32_16x16x4_f32(false, a, false, b1v, (short)0, c1, false, false);
            c2 = __builtin_amdgcn_wmma_f32_16x16x4_f32(false, a, false, b2v, (short)0, c2, false, false);
            c3 = __builtin_amdgcn_wmma_f32_16x16x4_f32(false, a, false, b3v, (short)0, c3, false, false);
        }

        // fused epilogue: write H, and AGG = H*dinv^2 + bias (self-loop term)
        const float bs0 = bias[n];
        const float bs1 = bias[n + 16];
        const float bs2 = bias[n + 32];
        const float bs3 = bias[n + 48];
        float* hrow = H   + (size_t)rowBase * HID;
        float* grow = AGG + (size_t)rowBase * HID;
#pragma unroll
        for (int j = 0; j < 8; ++j) {
            int r = j + 8 * half;                   // C/D layout: M = j + 8*(lane/16)
            float di = dinv[rowBase + r];
            float sc = di * di;
            size_t ro = (size_t)r * HID;
            hrow[ro + n]      = c0[j];  grow[ro + n]      = c0[j] * sc + bs0;
            hrow[ro + 16 + n] = c1[j];  grow[ro + 16 + n] = c1[j] * sc + bs1;
            hrow[ro + 32 + n] = c2[j];  grow[ro + 32 + n] = c2[j] * sc + bs2;
            hrow[ro + 48 + n] = c3[j];  grow[ro + 48 + n] = c3[j] * sc + bs3;
        }
    }
}

// ---------------- edge scatter: AGG[dst] += H[src] * dinv[src]*dinv[dst] ----------------
// 64 consecutive lanes handle the 64 features of one edge -> 256B contiguous
// gather and contiguous L2 atomics (H/AGG fit in the 192MB L2).
__global__ void k_scatter(const int* __restrict__ src, const int* __restrict__ dst,
                          const float* __restrict__ H, const float* __restrict__ dinv,
                          float* __restrict__ AGG) {
    int t = blockIdx.x * blockDim.x + threadIdx.x;  // E*64 = 51.2M < 2^31
    int e = t >> 6;
    int f = t & 63;
    if (e < N_EDGES) {
        int s = src[e], d = dst[e];
        float coef = dinv[s] * dinv[d];
        atomicAdd(&AGG[(size_t)d * HID + f], H[(size_t)s * HID + f] * coef);
    }
}

// ---------------- readout GEMM: out[N,10] = relu(A) @ Wr + br ----------------
// Wr (64x10) staged zero-padded to 16 cols in LDS (pair swizzle, stride 96:
// 96 mod 64 = 32 -> the two half-waves hit disjoint bank halves). No lane-varying
// control flow in the inner loop; B addresses are base + constant offsets.
__global__ __launch_bounds__(256) void k_gemm_out(const float* __restrict__ A,
                                                  const float* __restrict__ Wr,
                                                  const float* __restrict__ br,
                                                  float* __restrict__ out) {
    constexpr int PSTRIDE = 96;                     // 32 cols/pair-row + 64 pad
    __shared__ float Wl[(HID / 2) * PSTRIDE];       // 32 pair-rows

    for (int i = threadIdx.x; i < HID * 16; i += 256) {
        int k  = i >> 4;
        int nn = i & 15;
        float v = (nn < OUT_DIM) ? Wr[(size_t)k * OUT_DIM + nn] : 0.0f;
        Wl[(k >> 1) * PSTRIDE + (nn << 1) + (k & 1)] = v;
    }
    __syncthreads();

    const int lane   = threadIdx.x & 31;
    const int wave   = threadIdx.x >> 5;
    const int rowBlk = blockIdx.x * 8 + wave;
    if (rowBlk < ROW_BLOCKS) {                      // wave-uniform guard
        const int rowBase = rowBlk * 16;
        const int m    = lane & 15;
        const int n    = lane & 15;
        const int half = lane >> 4;
        const int kOff = half * 2;

        v8f c = {};
        const float* arow  = A + (size_t)(rowBase + m) * HID + kOff;
        const float* wbase = &Wl[half * PSTRIDE + 2 * n];

#pragma unroll
        for (int k0 = 0; k0 < HID; k0 += 4) {
            v2f a;
            a[0] = fmaxf(arow[k0], 0.0f);
            a[1] = fmaxf(arow[k0 + 1], 0.0f);
            v2f b = *(const v2f*)(wbase + (k0 >> 1) * PSTRIDE);
            c = __builtin_amdgcn_wmma_f32_16x16x4_f32(false, a, false, b, (short)0, c, false, false);
        }

        if (n < OUT_DIM) {                          // divergence only after all WMMAs
            float bias = br[n];
#pragma unroll
            for (int j = 0; j < 8; ++j) {
                int r = j + 8 * half;
                out[(size_t)(rowBase + r) * OUT_DIM + n] = c[j] + bias;
            }
        }
    }
}

// ---------------- row softmax in place (10-wide) ----------------
__global__ void k_softmax(float* out) {
    int r = blockIdx.x * blockDim.x + threadIdx.x;
    if (r >= N_NODES) return;
    float* row = out + (size_t)r * OUT_DIM;
    float mx = row[0];
#pragma unroll
    for (int j = 1; j < OUT_DIM; ++j) mx = fmaxf(mx, row[j]);
    float e[OUT_DIM];
    float s = 0.0f;
#pragma unroll
    for (int j = 0; j < OUT_DIM; ++j) { e[j] = expf(row[j] - mx); s += e[j]; }
    float inv = 1.0f / s;
#pragma unroll
    for (int j = 0; j < OUT_DIM; ++j) row[j] = e[j] * inv;
}

extern "C" void kernel_launch(void* const* d_in, const int* in_sizes, int n_in,
                              void* d_out, int out_size, void* d_ws, size_t ws_size,
                              hipStream_t stream) {
    const float* x   = (const float*)d_in[0];
    const int*   ei  = (const int*)  d_in[1];     // (2, E) row-major
    const int*   src = ei;
    const int*   dst = ei + N_EDGES;
    const float* W1  = (const float*)d_in[2];
    const float* b1  = (const float*)d_in[3];
    const float* W2  = (const float*)d_in[4];
    const float* b2  = (const float*)d_in[5];
    const float* W3  = (const float*)d_in[6];
    const float* b3  = (const float*)d_in[7];
    const float* Wr  = (const float*)d_in[8];
    const float* br  = (const float*)d_in[9];
    float* out = (float*)d_out;

    // workspace layout: dinv (200KB, padded) | H (12.8MB) | AGG (12.8MB)
    char* ws = (char*)d_ws;
    float* dinv = (float*)ws;
    float* H    = (float*)(ws + 256 * 1024);
    float* AGG  = (float*)(ws + 256 * 1024 + (size_t)N_NODES * HID * sizeof(float));

    const int T = 256;
    const int gNodes = (N_NODES + T - 1) / T;
    const int gEdges = (N_EDGES + T - 1) / T;
    const int gScat  = (N_EDGES * HID) / T;        // exact: 51.2M / 256
    const int gGemm  = (ROW_BLOCKS + 7) / 8;       // 8 waves per block

    // normalization coefficients
    k_init_deg<<<gNodes, T, 0, stream>>>(dinv);
    k_edge_deg<<<gEdges, T, 0, stream>>>(dst, dinv);
    k_rsqrt_inplace<<<gNodes, T, 0, stream>>>(dinv);

    // layer 1: GEMM (+fused self-loop/bias into AGG), then edge scatter
    k_gcn_gemm<IN_DIM, false><<<gGemm, T, 0, stream>>>(x, W1, b1, dinv, H, AGG);
    k_scatter<<<gScat, T, 0, stream>>>(src, dst, H, dinv, AGG);

    // layer 2 (relu of previous AGG folded into the A-load; in-place A==AGG is safe)
    k_gcn_gemm<HID, true><<<gGemm, T, 0, stream>>>(AGG, W2, b2, dinv, H, AGG);
    k_scatter<<<gScat, T, 0, stream>>>(src, dst, H, dinv, AGG);

    // layer 3
    k_gcn_gemm<HID, true><<<gGemm, T, 0, stream>>>(AGG, W3, b3, dinv, H, AGG);
    k_scatter<<<gScat, T, 0, stream>>>(src, dst, H, dinv, AGG);

    // readout + softmax
    k_gemm_out<<<gGemm, T, 0, stream>>>(AGG, Wr, br, out);
    k_softmax<<<gNodes, T, 0, stream>>>(out);
}